// SpatialDeformer3D_19069654794344
// MI455X (gfx1250) — compile-verified
//
#include <hip/hip_runtime.h>

// 3D spatial deformer (trilinear resample at base-grid + deformation).
// vol: (B,H,W,D,1) f32, deformation: (B,H,W,D,3) f32, out: (B,H,W,D,1) f32.
// B=2, H=W=D=160. Memory-bound gather kernel: volume stays resident in the
// 192MB L2; deformation/output are streamed with non-temporal hints.

namespace {
constexpr int S  = 160;
constexpr int S2 = S * S;    // 25600
constexpr int S3 = S * S * S;
constexpr float kLinStep = 2.0f / 159.0f;  // linspace(-1,1,160) step

// Faithful port of the reference coordinate handling:
//   c01 = 0.5*(c+1)*S ; i0 = floor ; i1 = i0+1 (pre-clip) ; clip both to [0,S-1]
//   w_lo = float(i1_clipped) - c01 ; w_hi = c01 - float(i0_clipped)
__device__ __forceinline__ void corner(float c, int& i0, int& i1,
                                       float& wlo, float& whi) {
  const float cf = 0.5f * (c + 1.0f) * (float)S;
  const int a0r = (int)floorf(cf);
  const int a1r = a0r + 1;
  const int a0 = min(max(a0r, 0), S - 1);
  const int a1 = min(max(a1r, 0), S - 1);
  wlo = (float)a1 - cf;   // weight paired with the *low* index gathers
  whi = cf - (float)a0;   // weight paired with the *high* index gathers
  i0 = a0;
  i1 = a1;
}
} // namespace

__global__ __launch_bounds__(256)
void deform_trilinear_kernel(const float* __restrict__ vol,
                             const float* __restrict__ def,
                             float* __restrict__ out) {
  // Lanes run along w so the 8 data-dependent gathers (x-fastest in the
  // reference's z*W*H + y*W + x flat index) are nearly unit-stride per wave.
  const int lane = threadIdx.x & 31;
  const int dt   = threadIdx.x >> 5;          // 0..7
  const int w = (blockIdx.x << 5) + lane;     // gridDim.x = 5
  const int d = (blockIdx.y << 3) + dt;       // gridDim.y = 20
  const int h = blockIdx.z % S;               // gridDim.z = 2*160
  const int b = blockIdx.z / S;

  // Output / deformation element index (C == 1, d fastest).
  const int base = ((b * S + h) * S + w) * S + d;

  // Deformation: streamed once -> non-temporal so it doesn't evict the volume.
  const float* dp = def + (size_t)base * 3;
  const float defx = __builtin_nontemporal_load(dp + 0);
  const float defy = __builtin_nontemporal_load(dp + 1);
  const float defz = __builtin_nontemporal_load(dp + 2);

  // meshgrid('xy'): x <- linspace over W at index w, y over H at h, z over D at d.
  const float xn = fmaf((float)w, kLinStep, -1.0f) + defx;
  const float yn = fmaf((float)h, kLinStep, -1.0f) + defy;
  const float zn = fmaf((float)d, kLinStep, -1.0f) + defz;

  int x0, x1, y0, y1, z0, z1;
  float wxl, wxh, wyl, wyh, wzl, wzh;
  corner(xn, x0, x1, wxl, wxh);
  corner(yn, y0, y1, wyl, wyh);
  corner(zn, z0, z1, wzl, wzh);

  // Reference gather: flat = z*(W*H) + y*W + x  (x fastest).
  const float* vb = vol + (size_t)b * S3;
  const int r_y0z0 = z0 * S2 + y0 * S;
  const int r_y1z0 = z0 * S2 + y1 * S;
  const int r_y0z1 = z1 * S2 + y0 * S;
  const int r_y1z1 = z1 * S2 + y1 * S;

  const float v000 = vb[r_y0z0 + x0];  // V(x0,y0,z0)
  const float v100 = vb[r_y0z0 + x1];  // V(x1,y0,z0)
  const float v010 = vb[r_y1z0 + x0];  // V(x0,y1,z0)
  const float v110 = vb[r_y1z0 + x1];  // V(x1,y1,z0)
  const float v001 = vb[r_y0z1 + x0];  // V(x0,y0,z1)
  const float v101 = vb[r_y0z1 + x1];  // V(x1,y0,z1)
  const float v011 = vb[r_y1z1 + x0];  // V(x0,y1,z1)
  const float v111 = vb[r_y1z1 + x1];  // V(x1,y1,z1)

  // Faithful reproduction of the reference's weight<->corner pairing quirk:
  // the z-style weight factor follows the *x* gather index and vice versa.
  float res;
  res = (wxl * wyl * wzl) * v000;               // (x1f-x)(y1f-y)(z1f-z) g(x0,y0,z0)
  res = fmaf(wxl * wyl * wzh, v100, res);       // (x1f-x)(y1f-y)(z-z0f) g(x1,y0,z0)
  res = fmaf(wxl * wyh * wzl, v010, res);       // (x1f-x)(y-y0f)(z1f-z) g(x0,y1,z0)
  res = fmaf(wxl * wyh * wzh, v110, res);       // (x1f-x)(y-y0f)(z-z0f) g(x1,y1,z0)
  res = fmaf(wxh * wyl * wzl, v001, res);       // (x-x0f)(y1f-y)(z1f-z) g(x0,y0,z1)
  res = fmaf(wxh * wyl * wzh, v101, res);       // (x-x0f)(y1f-y)(z-z0f) g(x1,y0,z1)
  res = fmaf(wxh * wyh * wzl, v011, res);       // (x-x0f)(y-y0f)(z1f-z) g(x0,y1,z1)
  res = fmaf(wxh * wyh * wzh, v111, res);       // (x-x0f)(y-y0f)(z-z0f) g(x1,y1,z1)

  // Output: written once -> non-temporal store (L2 merges the strided bytes).
  __builtin_nontemporal_store(res, out + base);
}

extern "C" void kernel_launch(void* const* d_in, const int* in_sizes, int n_in,
                              void* d_out, int out_size, void* d_ws, size_t ws_size,
                              hipStream_t stream) {
  const float* vol = (const float*)d_in[0];  // (2,160,160,160,1) f32
  const float* def = (const float*)d_in[1];  // (2,160,160,160,3) f32
  float* out = (float*)d_out;                // (2,160,160,160,1) f32

  dim3 grid(S / 32, S / 8, 2 * S);           // (5, 20, 320)
  deform_trilinear_kernel<<<grid, dim3(256), 0, stream>>>(vol, def, out);
}